// GCN_80479097192742
// MI455X (gfx1250) — compile-verified
//
#include <hip/hip_runtime.h>
#include <stdint.h>

// ---------------------------------------------------------------------------
// GCNConv (fp32 WMMA GEMM, 2x2 register tiling) + degree-normalized
// scatter-add + BN(eval) + ReLU.  Target: gfx1250 (CDNA5), wave32.
// ---------------------------------------------------------------------------

typedef float v2f __attribute__((ext_vector_type(2)));
typedef float v8f __attribute__((ext_vector_type(8)));

#define C_IN    512
#define C_HID   256
#define LDA     516     // LDS row stride (floats); 516 % 64 == 4 -> conflict-free A reads
#define MROWS   32      // rows per block (two 16-row WMMA M-tiles)
#define BN_EPSF 1e-5f

// ---- degree counting --------------------------------------------------------
__global__ __launch_bounds__(256) void k_zero_deg(uint32_t* __restrict__ deg, int n) {
  int i = blockIdx.x * 256 + threadIdx.x;
  if (i < n) deg[i] = 0u;
}

__global__ __launch_bounds__(256) void k_count_deg(const long long* __restrict__ ei,
                                                   uint32_t* __restrict__ deg, int E) {
  int e = blockIdx.x * 256 + threadIdx.x;
  if (e < E) atomicAdd(&deg[(int)ei[E + e]], 1u);
}

__global__ __launch_bounds__(256) void k_dinv(const uint32_t* __restrict__ deg,
                                              float* __restrict__ dinv, int n) {
  int i = blockIdx.x * 256 + threadIdx.x;
  if (i < n) dinv[i] = rsqrtf((float)deg[i] + 1.0f);   // +1 = self loop
}

// ---- fp32 WMMA GEMM: H[N,256] = X[N,512] * W[512,256] -----------------------
// Block: 256 threads (8 waves). Block covers rows [32b, 32b+32) x all 256 cols.
// Each wave owns columns [32w, 32w+32) and BOTH 16-row M-tiles: a 2x2 tile of
// 16x16 WMMA accumulators. One (b0,b1) pair feeds two WMMAs each -> 1.0
// wmma per global B load (vs 0.5 with a 1x2 tile), halving W/L2 traffic.
__global__ __launch_bounds__(256) void k_gemm_wmma(const float* __restrict__ X,
                                                   const float* __restrict__ W,
                                                   float* __restrict__ H) {
  __shared__ float As[MROWS * LDA];            // 32x516 floats = 66,048 B
  const int row0 = blockIdx.x * MROWS;
  const int tid  = threadIdx.x;

  // Cooperative load of the 32x512 fp32 A-tile (64 KB) into LDS, float4 wide.
  const float4* Xv = (const float4*)(X + (size_t)row0 * C_IN);
  #pragma unroll
  for (int i = 0; i < 16; ++i) {
    int idx = tid + i * 256;          // 0..4095 float4s
    int r   = idx >> 7;               // / 128
    int c4  = idx & 127;
    float4 v = Xv[r * 128 + c4];
    float* d = &As[r * LDA + c4 * 4];
    d[0] = v.x; d[1] = v.y; d[2] = v.z; d[3] = v.w;
  }
  __syncthreads();

  const int wave = tid >> 5;
  const int lane = tid & 31;
  const int m    = lane & 15;              // A row / D column index
  const int kh   = (lane >> 4) << 1;       // lanes 0-15 -> K={0,1}; 16-31 -> K={2,3}
  const int ncol = wave * 32 + m;          // base output column for N-tile 0

  v8f c00 = {}, c01 = {}, c10 = {}, c11 = {};
  const float* Arow0 = &As[m * LDA + kh];          // M-tile 0 (rows 0..15)
  const float* Arow1 = &As[(m + 16) * LDA + kh];   // M-tile 1 (rows 16..31)

  for (int k0 = 0; k0 < C_IN; k0 += 4) {
    // A 16x4 (ISA layout): VGPR0 = K{0|2}, VGPR1 = K{1|3}
    v2f a0; a0.x = Arow0[k0]; a0.y = Arow0[k0 + 1];
    v2f a1; a1.x = Arow1[k0]; a1.y = Arow1[k0 + 1];
    // B 4x16 (ISA layout): VGPR0 lanes0-15 K=0 / lanes16-31 K=2; VGPR1 K=1/K=3
    const float* Wp = W + (size_t)(k0 + kh) * C_HID + ncol;
    v2f b0, b1;
    b0.x = Wp[0];   b0.y = Wp[C_HID];
    b1.x = Wp[16];  b1.y = Wp[C_HID + 16];
    // (neg_a, A, neg_b, B, c_mod, C, reuse_a, reuse_b)
    c00 = __builtin_amdgcn_wmma_f32_16x16x4_f32(false, a0, false, b0, (short)0, c00, false, false);
    c01 = __builtin_amdgcn_wmma_f32_16x16x4_f32(false, a0, false, b1, (short)0, c01, false, false);
    c10 = __builtin_amdgcn_wmma_f32_16x16x4_f32(false, a1, false, b0, (short)0, c10, false, false);
    c11 = __builtin_amdgcn_wmma_f32_16x16x4_f32(false, a1, false, b1, (short)0, c11, false, false);
  }

  // D layout: VGPR r -> M = r (lanes 0-15) / r+8 (lanes 16-31); N = lane&15
  const int mb = row0 + ((lane >> 4) << 3);
  #pragma unroll
  for (int r = 0; r < 8; ++r) {
    float* h0 = H + (size_t)(mb + r) * C_HID + ncol;
    h0[0]  = c00[r];
    h0[16] = c01[r];
    float* h1 = H + (size_t)(mb + 16 + r) * C_HID + ncol;
    h1[0]  = c10[r];
    h1[16] = c11[r];
  }
}

// ---- out = bias + h * dinv^2 (self-loop term; re-inits accumulator), float4 -
__global__ __launch_bounds__(256) void k_init_out(const float4* __restrict__ H4,
                                                  const float4* __restrict__ bias4,
                                                  const float* __restrict__ dinv,
                                                  float4* __restrict__ out4, int total4) {
  int i = blockIdx.x * 256 + threadIdx.x;
  if (i >= total4) return;
  int row = i >> 6;                 // 64 float4s per row
  int c4  = i & 63;
  float di = dinv[row];
  float s  = di * di;
  float4 h = H4[i];
  float4 b = bias4[c4];
  float4 o;
  o.x = b.x + h.x * s;
  o.y = b.y + h.y * s;
  o.z = b.z + h.z * s;
  o.w = b.w + h.w * s;
  out4[i] = o;
}

// ---- edge scatter: one wave per edge, 8 channels per lane -------------------
__global__ __launch_bounds__(256) void k_scatter(const long long* __restrict__ ei,
                                                 const float* __restrict__ H,
                                                 const float* __restrict__ dinv,
                                                 float* __restrict__ out, int E) {
  long long gid = (long long)blockIdx.x * 256 + threadIdx.x;
  int e = (int)(gid >> 5);
  if (e >= E) return;
  int lane = threadIdx.x & 31;
  int src = (int)ei[e];
  int dst = (int)ei[E + e];
  float norm = dinv[src] * dinv[dst];

  const float4* hp = (const float4*)(H + (size_t)src * C_HID) + lane * 2;
  float4 v0 = hp[0];
  float4 v1 = hp[1];
  float* op = out + (size_t)dst * C_HID + lane * 8;
  // Hardware fp32 atomics (global_atomic_add_f32); accumulator lives in L2.
  unsafeAtomicAdd(op + 0, v0.x * norm);
  unsafeAtomicAdd(op + 1, v0.y * norm);
  unsafeAtomicAdd(op + 2, v0.z * norm);
  unsafeAtomicAdd(op + 3, v0.w * norm);
  unsafeAtomicAdd(op + 4, v1.x * norm);
  unsafeAtomicAdd(op + 5, v1.y * norm);
  unsafeAtomicAdd(op + 6, v1.z * norm);
  unsafeAtomicAdd(op + 7, v1.w * norm);
}

// ---- BatchNorm1d (eval) + ReLU, in place, float4 ----------------------------
__global__ __launch_bounds__(256) void k_bn_relu(float4* __restrict__ out4,
                                                 const float4* __restrict__ gamma4,
                                                 const float4* __restrict__ beta4,
                                                 const float4* __restrict__ mean4,
                                                 const float4* __restrict__ var4, int total4) {
  int i = blockIdx.x * 256 + threadIdx.x;
  if (i >= total4) return;
  int c4 = i & 63;
  float4 v  = out4[i];
  float4 mu = mean4[c4];
  float4 vr = var4[c4];
  float4 g  = gamma4[c4];
  float4 bt = beta4[c4];
  float4 y;
  y.x = fmaxf((v.x - mu.x) * rsqrtf(vr.x + BN_EPSF) * g.x + bt.x, 0.0f);
  y.y = fmaxf((v.y - mu.y) * rsqrtf(vr.y + BN_EPSF) * g.y + bt.y, 0.0f);
  y.z = fmaxf((v.z - mu.z) * rsqrtf(vr.z + BN_EPSF) * g.z + bt.z, 0.0f);
  y.w = fmaxf((v.w - mu.w) * rsqrtf(vr.w + BN_EPSF) * g.w + bt.w, 0.0f);
  out4[i] = y;
}

// ---------------------------------------------------------------------------
extern "C" void kernel_launch(void* const* d_in, const int* in_sizes, int n_in,
                              void* d_out, int out_size, void* d_ws, size_t ws_size,
                              hipStream_t stream) {
  const float*     x     = (const float*)d_in[0];
  const long long* ei    = (const long long*)d_in[1];   // int64 edge_index [2,E]
  const float*     W     = (const float*)d_in[2];
  const float*     b     = (const float*)d_in[3];
  const float*     gamma = (const float*)d_in[4];
  const float*     beta  = (const float*)d_in[5];
  const float*     rmean = (const float*)d_in[6];
  const float*     rvar  = (const float*)d_in[7];

  const int Nn = in_sizes[0] / C_IN;     // 100000
  const int E  = in_sizes[1] / 2;        // 1600000
  float* out = (float*)d_out;

  // Workspace layout: H [N*256 f32] | deg [N u32] | dinv [N f32]
  char*     ws   = (char*)d_ws;
  float*    H    = (float*)ws;
  size_t    hby  = (size_t)Nn * C_HID * sizeof(float);
  uint32_t* deg  = (uint32_t*)(ws + hby);
  float*    dinv = (float*)(ws + hby + (size_t)Nn * sizeof(uint32_t));

  const int total4 = Nn * (C_HID / 4);

  k_zero_deg <<<(Nn + 255) / 256, 256, 0, stream>>>(deg, Nn);
  k_count_deg<<<(E  + 255) / 256, 256, 0, stream>>>(ei, deg, E);
  k_dinv     <<<(Nn + 255) / 256, 256, 0, stream>>>(deg, dinv, Nn);

  k_gemm_wmma<<<Nn / MROWS, 256, 0, stream>>>(x, W, H);   // N divisible by 32

  k_init_out <<<(total4 + 255) / 256, 256, 0, stream>>>((const float4*)H, (const float4*)b,
                                                        dinv, (float4*)out, total4);

  long long sthreads = (long long)E * 32;
  k_scatter  <<<(int)((sthreads + 255) / 256), 256, 0, stream>>>(ei, H, dinv, out, E);

  k_bn_relu  <<<(total4 + 255) / 256, 256, 0, stream>>>((float4*)out, (const float4*)gamma,
                                                        (const float4*)beta, (const float4*)rmean,
                                                        (const float4*)rvar, total4);
}